// GraphAttentionLayer_54717883351090
// MI455X (gfx1250) — compile-verified
//
#include <hip/hip_runtime.h>
#include <hip/hip_bf16.h>

// ---------------------------------------------------------------------------
// GAT layer, restructured: edge-level GEMMs algebraically folded into one
// node-level fused GEMM [N x 128] @ [128 x 640] done with V_WMMA_F32_16X16X4_F32,
// then a streaming edge pass (L2-resident gathers + f32 atomics).
// ---------------------------------------------------------------------------

typedef __attribute__((ext_vector_type(2))) float v2f;
typedef __attribute__((ext_vector_type(8))) float v8f;

#define F_IN   128
#define F_OUT  64
#define KHEADS 4
#define COLS   640            // 4*64 (U) | 4*64 (V) | 64 (m_self) | 64 (m_nb)
#define LDA    132            // padded LDS row stride (bank-conflict-free)

// ---------------------------------------------------------------------------
// Zero-fill workspace accumulators (harness poisons ws with 0xAA).
// ---------------------------------------------------------------------------
__global__ void zero_ws_kernel(float* __restrict__ p, size_t n) {
    size_t i = (size_t)blockIdx.x * blockDim.x + threadIdx.x;
    if (i < n) p[i] = 0.0f;
}

// ---------------------------------------------------------------------------
// Fold Wt@Wa_top (+bt@Wa_top+ba) and Wt@Wa_bot (+bt@Wa_bot) plus Wv/Wn into a
// single [128 x 640] weight matrix Wbig and [640] bias bbig.
//   cols [  0,256): U head h = col/64       (dst side, bias includes ba)
//   cols [256,512): V head h = (col-256)/64 (src side)
//   cols [512,576): Wv / bv  (m_self)
//   cols [576,640): Wn / bn  (m_nb)
// ---------------------------------------------------------------------------
__global__ void build_weights_kernel(const float* __restrict__ Wv, const float* __restrict__ bv,
                                     const float* __restrict__ Wn, const float* __restrict__ bn,
                                     const float* __restrict__ Wt, const float* __restrict__ bt,
                                     const float* __restrict__ Wa, const float* __restrict__ ba,
                                     float* __restrict__ Wbig, float* __restrict__ bbig) {
    const int WTOT = F_IN * COLS;  // 81920
    int idx = blockIdx.x * blockDim.x + threadIdx.x;
    if (idx < WTOT) {
        int k = idx / COLS, col = idx % COLS;
        float val;
        if (col < 512) {
            int side = col >> 8;          // 0 = top (dst/u), 1 = bottom (src/v)
            int hc = col & 255;
            int h = hc >> 6, c = hc & 63;
            const float* wt = Wt + (size_t)h * F_IN * F_OUT + (size_t)k * F_OUT; // Wt[h][k][:]
            const float* wa = Wa + (size_t)h * 2 * F_OUT * F_OUT + (size_t)side * F_OUT * F_OUT + c;
            float s = 0.0f;
            #pragma unroll 8
            for (int j = 0; j < F_OUT; ++j) s += wt[j] * wa[(size_t)j * F_OUT];
            val = s;
        } else if (col < 576) {
            val = Wv[(size_t)k * F_OUT + (col - 512)];
        } else {
            val = Wn[(size_t)k * F_OUT + (col - 576)];
        }
        Wbig[idx] = val;
    } else if (idx < WTOT + COLS) {
        int col = idx - WTOT;
        float val;
        if (col < 512) {
            int side = col >> 8;
            int hc = col & 255;
            int h = hc >> 6, c = hc & 63;
            const float* btp = bt + (size_t)h * F_OUT;
            const float* wa  = Wa + (size_t)h * 2 * F_OUT * F_OUT + (size_t)side * F_OUT * F_OUT + c;
            float s = 0.0f;
            #pragma unroll 8
            for (int j = 0; j < F_OUT; ++j) s += btp[j] * wa[(size_t)j * F_OUT];
            if (side == 0) s += ba[(size_t)h * F_OUT + c];   // ba added once per head
            val = s;
        } else if (col < 576) {
            val = bv[col - 512];
        } else {
            val = bn[col - 576];
        }
        bbig[col] = val;
    }
}

// ---------------------------------------------------------------------------
// Fused node GEMM:  O[N x 640] = X[N x 128] @ Wbig[128 x 640] + bbig
// Block = 160 threads = 5 waves; each wave computes one 16x16 tile with
// 32 x V_WMMA_F32_16X16X4_F32 over K=128. X tiles + weight tile staged in LDS.
//
// Fragment layouts per ISA 7.12.2 (f32, wave32):
//   A 16x4 : lane L(0..15): a.x=A[M=L][k0], a.y=A[L][k0+1]; lanes 16..31: K+2
//   B 4x16 : lane L(0..15): b.x=B[k0][N=L], b.y=B[k0+1][L]; lanes 16..31: K+2
//   C/D    : acc[r] = D[r + 8*(lane>=16)][lane&15]
// ---------------------------------------------------------------------------
__global__ __launch_bounds__(160) void node_gemm_kernel(const float* __restrict__ X,
                                                        const float* __restrict__ Wbig,
                                                        const float* __restrict__ bbig,
                                                        float* __restrict__ O,
                                                        int nNodes) {
    __shared__ float sA[5][16 * LDA];    // 5 M-tiles of X, padded rows
    __shared__ float sB[F_IN * 16];      // one 128x16 weight tile, layout [k][n]

    const int nTile  = blockIdx.x % (COLS / 16);   // 0..39
    const int mGroup = blockIdx.x / (COLS / 16);
    const int m0 = mGroup * 80;                    // 5 tiles * 16 rows
    const int n0 = nTile * 16;
    const int tid = threadIdx.x;

    // Stage weight tile (2048 floats). Wbig is 320 KB -> L2 resident.
    for (int i = tid; i < F_IN * 16; i += 160) {
        int k = i >> 4, n = i & 15;
        sB[i] = Wbig[(size_t)k * COLS + n0 + n];
    }
    // Stage 80 rows of X (coalesced float4 row loads -> conflict-free LDS).
    for (int i = tid; i < 80 * 32; i += 160) {
        int r = i >> 5, c4 = (i & 31) << 2;
        int gr = m0 + r; if (gr >= nNodes) gr = nNodes - 1;
        const float4 v = *(const float4*)(X + (size_t)gr * F_IN + c4);
        float* dp = &sA[r >> 4][(r & 15) * LDA + c4];
        dp[0] = v.x; dp[1] = v.y; dp[2] = v.z; dp[3] = v.w;
    }
    __syncthreads();

    const int wave  = tid >> 5;          // 0..4 -> which M tile
    const int lane  = tid & 31;
    const int lrow  = lane & 15;         // M (for A) / N (for B,D)
    const int khalf = (lane >> 4) << 1;  // 0 or 2: K sub-offset for upper lanes

    const float bcol = bbig[n0 + lrow];  // bias depends only on column
    v8f acc = { bcol, bcol, bcol, bcol, bcol, bcol, bcol, bcol };

    const float* aBase = &sA[wave][lrow * LDA + khalf];
    const float* bBase = &sB[khalf * 16 + lrow];

    #pragma unroll
    for (int k0 = 0; k0 < F_IN; k0 += 4) {
        float2 av = *(const float2*)(aBase + k0);       // ds_load_b64
        v2f a; a.x = av.x; a.y = av.y;
        v2f b; b.x = bBase[k0 * 16]; b.y = bBase[k0 * 16 + 16];
        acc = __builtin_amdgcn_wmma_f32_16x16x4_f32(
            /*neg_a=*/false, a, /*neg_b=*/false, b,
            /*c_mod=*/(short)0, acc, /*reuse_a=*/false, /*reuse_b=*/false);
    }

    const int rbase = m0 + wave * 16 + ((lane >> 4) << 3);
    const int col   = n0 + lrow;
    #pragma unroll
    for (int r = 0; r < 8; ++r) {
        int gr = rbase + r;
        if (gr < nNodes) O[(size_t)gr * COLS + col] = acc[r];
    }
}

// ---------------------------------------------------------------------------
// Edge pass: one wave per edge, 2 channels per lane.
//   att = 0.25 * sum_h relu(U_h[dst] + V_h[src]);  e = exp(att)
//   atomically accumulate e into denom[dst] and e*m_nb[src] into num[dst].
// O rows (128 MB) are L2-resident -> random gathers served by the 192 MB L2.
// ---------------------------------------------------------------------------
__global__ __launch_bounds__(256) void edge_pass_kernel(const int* __restrict__ src,
                                                        const int* __restrict__ dst,
                                                        const float* __restrict__ O,
                                                        float* __restrict__ denom,
                                                        float* __restrict__ num,
                                                        int nEdges) {
    int e = blockIdx.x * 8 + (threadIdx.x >> 5);
    if (e >= nEdges) return;
    const int lane = threadIdx.x & 31;
    const int s = src[e], d = dst[e];
    const float* Rd = O + (size_t)d * COLS;          // U at cols [0,256)
    const float* Rs = O + (size_t)s * COLS;          // V at cols [256,512), m_nb at [576,640)

    float ax = 0.0f, ay = 0.0f;
    #pragma unroll
    for (int h = 0; h < KHEADS; ++h) {
        float2 u = *(const float2*)(Rd + h * F_OUT + 2 * lane);
        float2 v = *(const float2*)(Rs + 256 + h * F_OUT + 2 * lane);
        ax += fmaxf(u.x + v.x, 0.0f);
        ay += fmaxf(u.y + v.y, 0.0f);
    }
    const float ex = __expf(ax * 0.25f);
    const float ey = __expf(ay * 0.25f);
    const float2 m = *(const float2*)(Rs + 576 + 2 * lane);

    const size_t o = (size_t)d * F_OUT + 2 * lane;
    unsafeAtomicAdd(denom + o,     ex);
    unsafeAtomicAdd(denom + o + 1, ey);
    unsafeAtomicAdd(num   + o,     ex * m.x);
    unsafeAtomicAdd(num   + o + 1, ey * m.y);
}

// ---------------------------------------------------------------------------
// Finalize per (node, channel):
//   att_self = 0.25 * sum_h relu(U_h[n] + V_h[n]);  es = exp(att_self)
//   out = relu((es*m_self + num) / (es + denom))
// ---------------------------------------------------------------------------
__global__ __launch_bounds__(256) void finalize_kernel(const float* __restrict__ O,
                                                       const float* __restrict__ denom,
                                                       const float* __restrict__ num,
                                                       float* __restrict__ out,
                                                       int nNodes) {
    int idx = blockIdx.x * blockDim.x + threadIdx.x;
    if (idx >= nNodes * F_OUT) return;
    const int n = idx >> 6, c = idx & 63;
    const float* row = O + (size_t)n * COLS;
    float a = 0.0f;
    #pragma unroll
    for (int h = 0; h < KHEADS; ++h)
        a += fmaxf(row[h * F_OUT + c] + row[256 + h * F_OUT + c], 0.0f);
    const float es  = __expf(a * 0.25f);
    const float den = es + denom[idx];
    const float nm  = es * row[512 + c] + num[idx];
    out[idx] = fmaxf(nm / den, 0.0f);
}

// ---------------------------------------------------------------------------
// Launch
// ---------------------------------------------------------------------------
extern "C" void kernel_launch(void* const* d_in, const int* in_sizes, int n_in,
                              void* d_out, int out_size, void* d_ws, size_t ws_size,
                              hipStream_t stream) {
    const float* x   = (const float*)d_in[0];
    const int*   src = (const int*)  d_in[1];
    const int*   dst = (const int*)  d_in[2];
    const float* Wv  = (const float*)d_in[3];
    const float* bv  = (const float*)d_in[4];
    const float* Wn  = (const float*)d_in[5];
    const float* bn  = (const float*)d_in[6];
    const float* Wt  = (const float*)d_in[7];
    const float* bt  = (const float*)d_in[8];
    const float* Wa  = (const float*)d_in[9];
    const float* ba  = (const float*)d_in[10];

    const int nNodes = in_sizes[0] / F_IN;
    const int nEdges = in_sizes[1];

    // Workspace layout (floats):
    float* Wbig  = (float*)d_ws;                        // 128*640
    float* bbig  = Wbig + F_IN * COLS;                  // 640
    float* O     = bbig + COLS;                         // nNodes*640
    float* denom = O + (size_t)nNodes * COLS;           // nNodes*64
    float* num   = denom + (size_t)nNodes * F_OUT;      // nNodes*64

    // 1) zero accumulators
    const size_t zn = (size_t)nNodes * (2 * F_OUT);
    zero_ws_kernel<<<(unsigned)((zn + 255) / 256), 256, 0, stream>>>(denom, zn);

    // 2) fold weights
    const int wtot = F_IN * COLS + COLS;
    build_weights_kernel<<<(wtot + 255) / 256, 256, 0, stream>>>(
        Wv, bv, Wn, bn, Wt, bt, Wa, ba, Wbig, bbig);

    // 3) fused node GEMM (WMMA)
    const int mTiles  = (nNodes + 15) / 16;
    const int mGroups = (mTiles + 4) / 5;
    node_gemm_kernel<<<mGroups * (COLS / 16), 160, 0, stream>>>(x, Wbig, bbig, O, nNodes);

    // 4) edge pass (gather + atomics)
    edge_pass_kernel<<<(nEdges + 7) / 8, 256, 0, stream>>>(src, dst, O, denom, num, nEdges);

    // 5) finalize
    const int outElems = nNodes * F_OUT;
    finalize_kernel<<<(outElems + 255) / 256, 256, 0, stream>>>(O, denom, num, (float*)d_out, nNodes);
}